// E3ResidualBlock_29609504538682
// MI455X (gfx1250) — compile-verified
//
#include <hip/hip_runtime.h>
#include <hip/hip_bf16.h>

typedef float v2f __attribute__((ext_vector_type(2)));
typedef float v8f __attribute__((ext_vector_type(8)));

// LDS layout strides (padded for bank-conflict avoidance on 64 banks)
constexpr int ZS  = 580;   // feature row stride (16 rows x 576 floats, pad to 580, mult of 4 for float4)
constexpr int WS  = 65;    // W row stride (64 + 1)
constexpr int WLS = 64 * WS; // per-l W matrix stride in LDS

struct MixP { float a0, a1, a2, bS, bM, cS, cP, cQ, bias; };

__device__ __forceinline__ float silu_f(float x) {
    return x * (1.0f / (1.0f + __expf(-x)));
}

// Fused scalar-activation + CG self tensor product for one (n,c) 9-vector.
// v holds [s | u0 u1 u2 | q0..q4] in e3nn real-SH order (m = -l..l).
__device__ __forceinline__ void tp_selfmix(float v[9], const MixP& w,
                                           float k0, float k1v, float k2v) {
    const float IS3  = 0.5773502691896258f;   // 1/sqrt(3)
    const float IS5  = 0.4472135954999579f;   // 1/sqrt(5)
    const float S121 = 0.5477225575051661f;   // sqrt(3/10)
    const float T121 = 0.31622776601683794f;  // 1/sqrt(10)
    const float R2   = 1.4142135623730951f;   // sqrt(2)
    const float IS6  = 0.40824829046386296f;  // 1/sqrt(6)
    const float IR2  = 0.7071067811865476f;   // 1/sqrt(2)
    const float C1   = 1.0690449676496976f;   // 4/sqrt(14)
    const float C2   = 0.9258200997725514f;   // 2*sqrt(3)/sqrt(14)
    const float C3   = 0.5345224838248488f;   // 2/sqrt(14)
    const float C4   = 0.4629100498862757f;   // sqrt(3)/sqrt(14)
    const float C5   = 0.2672612419124244f;   // 1/sqrt(14)

    float s  = silu_f(v[0]);
    float u0 = v[1], u1 = v[2], u2 = v[3];
    float q0 = v[4], q1 = v[5], q2 = v[6], q3 = v[7], q4 = v[8];

    float uu = u0*u0 + u1*u1 + u2*u2;
    float qq = q0*q0 + q1*q1 + q2*q2 + q3*q3 + q4*q4;
    // l3 = 0 : paths (0,0,0), (1,1,0)->-uu/sqrt3, (2,2,0)->+qq/sqrt5 ; /sqrt(3) ; +b ; +k*s
    v[0] = (w.a0 * s * s - w.a1 * uu * IS3 + w.a2 * qq * IS5) * IS3 + w.bias + k0 * s;

    // l3 = 1 : (0,1,1)==(1,0,1)=s*u ; (1,2,1)==(2,1,1)= -sqrt(3/5) * S(q) u
    float M0 = -S121 * (q1*u1 + q0*u2) + (T121*q2 + S121*q4) * u0;
    float M1 = -S121 * (q1*u0 + q3*u2) - 2.0f * T121 * q2 * u1;
    float M2 = -S121 * (q3*u1 + q0*u0) + (T121*q2 - S121*q4) * u2;
    float su = w.bS * s;
    v[1] = (su*u0 + w.bM*M0) * 0.5f + k1v * u0;
    v[2] = (su*u1 + w.bM*M1) * 0.5f + k1v * u1;
    v[3] = (su*u2 + w.bM*M2) * 0.5f + k1v * u2;

    // l3 = 2 : (0,2,2)==(2,0,2)=s*q ; (1,1,2)=P(u,u) ; (2,2,2)=Q(q,q)
    float Pc0 = R2 * u2 * u0;
    float Pc1 = R2 * u0 * u1;
    float Pc2 = (2.f*u1*u1 - u2*u2 - u0*u0) * IS6;
    float Pc3 = R2 * u2 * u1;
    float Pc4 = (u2*u2 - u0*u0) * IR2;

    float Q0 =  C1*q0*q2 - C2*q1*q3;
    float Q1 = -C2*q0*q3 + C2*q1*q4 - C3*q1*q2;
    float Q2 =  C5 * (2.f*q0*q0 - q1*q1 - 2.f*q2*q2 - q3*q3 + 2.f*q4*q4);
    float Q3 = -C2*(q3*q4 + q0*q1) - C3*q2*q3;
    float Q4 =  C1*q2*q4 - C4*(q3*q3 - q1*q1);

    float sq = w.cS * s;
    v[4] = (sq*q0 + w.cP*Pc0 + w.cQ*Q0) * 0.5f + k2v * q0;
    v[5] = (sq*q1 + w.cP*Pc1 + w.cQ*Q1) * 0.5f + k2v * q1;
    v[6] = (sq*q2 + w.cP*Pc2 + w.cQ*Q2) * 0.5f + k2v * q2;
    v[7] = (sq*q3 + w.cP*Pc3 + w.cQ*Q3) * 0.5f + k2v * q3;
    v[8] = (sq*q4 + w.cP*Pc4 + w.cQ*Q4) * 0.5f + k2v * q4;
}

// Channel mix: y[n,d,comp] = sum_c z[n,c,comp] * (W[l]/8)[c,d]  (+bias on comp 0)
// 36 output tiles (9 comps x 4 col-tiles of 16 ch) round-robined over 8 waves.
// Each tile: 16 chained v_wmma_f32_16x16x4_f32 (K = 64).
__device__ __forceinline__ void channel_mix16(const float* __restrict__ zs,
                                              float* __restrict__ yd,
                                              const float* __restrict__ wl,
                                              const float* __restrict__ bm,
                                              int tid) {
    const int wave = tid >> 5;
    const int lane = tid & 31;
    const int M    = lane & 15;   // A row / B col (N)
    const int hi   = lane >> 4;   // half-wave select

    for (int t = wave; t < 36; t += 8) {
        const int comp = t >> 2;
        const int dt   = t & 3;
        const int l    = (comp == 0) ? 0 : ((comp < 4) ? 1 : 2);
        const int d    = dt * 16 + M;
        const float* wp = wl + l * WLS + d;
        const float* zp = zs + M * ZS + comp;

        v8f acc = {0.f, 0.f, 0.f, 0.f, 0.f, 0.f, 0.f, 0.f};
#pragma unroll
        for (int kk = 0; kk < 16; ++kk) {
            const int c0 = kk * 4 + hi * 2;   // lanes 16-31 carry K+2 per ISA A layout
            v2f av, bv;
            av.x = zp[c0 * 9];        // A[M][c0]
            av.y = zp[c0 * 9 + 9];    // A[M][c0+1]
            bv.x = wp[c0 * WS];       // B[c0][d]
            bv.y = wp[c0 * WS + WS];  // B[c0+1][d]
            acc = __builtin_amdgcn_wmma_f32_16x16x4_f32(
                false, av, false, bv, (short)0, acc, false, false);
        }
        const float bias = (comp == 0) ? bm[d] : 0.0f;
#pragma unroll
        for (int i = 0; i < 8; ++i) {
            const int row = i + hi * 8;       // D: VGPR i -> M=i (lo half), M=i+8 (hi half)
            yd[row * ZS + d * 9 + comp] = acc[i] + bias;
        }
    }
}

__global__ __launch_bounds__(256) void e3_residual_block_kernel(
    const float* __restrict__ x,   const float* __restrict__ gamma,
    const float* __restrict__ w10, const float* __restrict__ w11,
    const float* __restrict__ w12, const float* __restrict__ b1,
    const float* __restrict__ k1,  const float* __restrict__ Wm1,
    const float* __restrict__ bm1, const float* __restrict__ w20,
    const float* __restrict__ w21, const float* __restrict__ w22,
    const float* __restrict__ b2,  const float* __restrict__ k2,
    const float* __restrict__ Wm2, const float* __restrict__ bm2,
    float* __restrict__ out) {
    __shared__ __align__(16) float zbuf[16 * ZS];
    __shared__ __align__(16) float ybuf[16 * ZS];
    __shared__ __align__(16) float wbuf[3 * WLS];
    __shared__ float msbuf[48];

    const int tid = threadIdx.x;
    const size_t base = (size_t)blockIdx.x * 9216;  // 16 rows * 64 ch * 9
    const float* xg = x + base;

    // Stage W1/sqrt(C) into LDS (padded rows) + stage x tile via float4.
    for (int e = 0; e < 48; ++e) {
        int g = e * 256 + tid;
        wbuf[(g >> 12) * WLS + ((g >> 6) & 63) * WS + (g & 63)] = Wm1[g] * 0.125f;
    }
    for (int e = 0; e < 9; ++e) {
        int j4 = e * 256 + tid;
        int n = j4 / 144, r4 = j4 - n * 144;
        reinterpret_cast<float4*>(zbuf + n * ZS)[r4] =
            reinterpret_cast<const float4*>(xg)[j4];
    }
    __syncthreads();

    // Per-n RMS norms per l (reduce over 64 ch x (2l+1)); 16 lanes per n via half-wave shuffles.
    {
        const int g = tid >> 4, lg = tid & 15;
        float s0 = 0.f, s1 = 0.f, s2 = 0.f;
        for (int jj = 0; jj < 4; ++jj) {
            const float* p = zbuf + g * ZS + (lg + jj * 16) * 9;
            float a = p[0]; s0 += a * a;
            s1 += p[1]*p[1] + p[2]*p[2] + p[3]*p[3];
            s2 += p[4]*p[4] + p[5]*p[5] + p[6]*p[6] + p[7]*p[7] + p[8]*p[8];
        }
        for (int o = 1; o < 16; o <<= 1) {
            s0 += __shfl_xor(s0, o, 16);
            s1 += __shfl_xor(s1, o, 16);
            s2 += __shfl_xor(s2, o, 16);
        }
        if (lg == 0) {
            msbuf[g * 3 + 0] = rsqrtf(s0 * (1.f / 64.f)  + 1e-6f);
            msbuf[g * 3 + 1] = rsqrtf(s1 * (1.f / 192.f) + 1e-6f);
            msbuf[g * 3 + 2] = rsqrtf(s2 * (1.f / 320.f) + 1e-6f);
        }
    }
    __syncthreads();

    const MixP P1 = { w10[0], w10[1], w10[2], w11[0] + w11[1], w11[2] + w11[3],
                      w12[0] + w12[1], w12[2], w12[3], b1[0] };
    const MixP P2 = { w20[0], w20[1], w20[2], w21[0] + w21[1], w21[2] + w21[3],
                      w22[0] + w22[1], w22[2], w22[3], b2[0] };

    // Stage A: norm * gamma -> SiLU -> selfmix1 (in place in zbuf), 4 (n,c) pairs/thread.
    for (int e = 0; e < 4; ++e) {
        const int p = e * 256 + tid;
        const int n = p >> 6, c = p & 63;
        float* vp = zbuf + n * ZS + c * 9;
        const float i0 = msbuf[n * 3], i1 = msbuf[n * 3 + 1], i2 = msbuf[n * 3 + 2];
        const float g0 = gamma[c * 3] * i0, g1 = gamma[c * 3 + 1] * i1, g2 = gamma[c * 3 + 2] * i2;
        float v[9];
        v[0] = vp[0] * g0;
        v[1] = vp[1] * g1; v[2] = vp[2] * g1; v[3] = vp[3] * g1;
        v[4] = vp[4] * g2; v[5] = vp[5] * g2; v[6] = vp[6] * g2;
        v[7] = vp[7] * g2; v[8] = vp[8] * g2;
        tp_selfmix(v, P1, k1[c * 3], k1[c * 3 + 1], k1[c * 3 + 2]);
#pragma unroll
        for (int m = 0; m < 9; ++m) vp[m] = v[m];
    }
    __syncthreads();

    channel_mix16(zbuf, ybuf, wbuf, bm1, tid);   // mix1 -> ybuf
    __syncthreads();

    // Re-stage W2/sqrt(C); Stage B: SiLU -> selfmix2 (ybuf -> zbuf).
    for (int e = 0; e < 48; ++e) {
        int g = e * 256 + tid;
        wbuf[(g >> 12) * WLS + ((g >> 6) & 63) * WS + (g & 63)] = Wm2[g] * 0.125f;
    }
    for (int e = 0; e < 4; ++e) {
        const int p = e * 256 + tid;
        const int n = p >> 6, c = p & 63;
        const float* sp = ybuf + n * ZS + c * 9;
        float v[9];
#pragma unroll
        for (int m = 0; m < 9; ++m) v[m] = sp[m];
        tp_selfmix(v, P2, k2[c * 3], k2[c * 3 + 1], k2[c * 3 + 2]);
        float* dp = zbuf + n * ZS + c * 9;
#pragma unroll
        for (int m = 0; m < 9; ++m) dp[m] = v[m];
    }
    __syncthreads();

    channel_mix16(zbuf, ybuf, wbuf, bm2, tid);   // mix2 -> ybuf
    __syncthreads();

    // Residual + streaming writeback (float4).
    float* og = out + base;
    for (int e = 0; e < 9; ++e) {
        int j4 = e * 256 + tid;
        int n = j4 / 144, r4 = j4 - n * 144;
        float4 yv = reinterpret_cast<const float4*>(ybuf + n * ZS)[r4];
        float4 xv = reinterpret_cast<const float4*>(xg)[j4];
        float4 r;
        r.x = yv.x + xv.x; r.y = yv.y + xv.y;
        r.z = yv.z + xv.z; r.w = yv.w + xv.w;
        reinterpret_cast<float4*>(og)[j4] = r;
    }
}

extern "C" void kernel_launch(void* const* d_in, const int* in_sizes, int n_in,
                              void* d_out, int out_size, void* d_ws, size_t ws_size,
                              hipStream_t stream) {
    const float* x     = (const float*)d_in[0];
    const float* gamma = (const float*)d_in[1];
    const float* w10   = (const float*)d_in[2];
    const float* w11   = (const float*)d_in[3];
    const float* w12   = (const float*)d_in[4];
    const float* b1    = (const float*)d_in[5];
    const float* k1    = (const float*)d_in[6];
    const float* Wm1   = (const float*)d_in[7];
    const float* bm1   = (const float*)d_in[8];
    const float* w20   = (const float*)d_in[9];
    const float* w21   = (const float*)d_in[10];
    const float* w22   = (const float*)d_in[11];
    const float* b2    = (const float*)d_in[12];
    const float* k2    = (const float*)d_in[13];
    const float* Wm2   = (const float*)d_in[14];
    const float* bm2   = (const float*)d_in[15];
    float* out = (float*)d_out;

    const int N = in_sizes[0] / 576;       // 32768
    const int blocks = N / 16;             // 2048 workgroups of 8 waves
    e3_residual_block_kernel<<<blocks, 256, 0, stream>>>(
        x, gamma, w10, w11, w12, b1, k1, Wm1, bm1,
        w20, w21, w22, b2, k2, Wm2, bm2, out);
}